// MultiHeadSpectralAttention_71476845740628
// MI455X (gfx1250) — compile-verified
//
#include <hip/hip_runtime.h>
#include <hip/hip_bf16.h>

typedef float v2f __attribute__((ext_vector_type(2)));
typedef float v8f __attribute__((ext_vector_type(8)));

#define NB     32
#define DDIM   256
#define HEADS  8
#define HD     32
#define NN     2304            // 48*48
#define FB     1153            // NN/2 + 1
#define KP     1168            // FB padded to 73*16
#define ADIM   18448           // HEADS*FB*2
#define RSQN   (1.0f/48.0f)    // 1/sqrt(2304)
#define TWO_PI 6.28318530717958647692f

// ---- workspace layout (float offsets) ----
#define SZ_TAB   (KP * NN)                       // 2,691,072
#define OFF_CF   0
#define OFF_SF   (OFF_CF + SZ_TAB)
#define OFF_CI   (OFF_SF + SZ_TAB)
#define OFF_SI   (OFF_CI + SZ_TAB)
#define SZ_F     (NB * HEADS * KP * HD)          // 9,568,256
#define OFF_FR   (OFF_SI + SZ_TAB)
#define OFF_FI   (OFF_FR + SZ_F)
#define OFF_CTX  (OFF_FI + SZ_F)                 // 8192
#define OFF_HMID (OFF_CTX + NB * DDIM)           // 8192
#define OFF_ADPT (OFF_HMID + NB * DDIM)          // 590,336
// total ~30.5M floats ~= 122 MB

__device__ __forceinline__ float gelu_exact(float v) {
    return 0.5f * v * (1.0f + erff(v * 0.70710678118654752f));
}

// D(16x16) = A(16x4,f32) * B(4x16,f32) + C  -- V_WMMA_F32_16X16X4_F32
__device__ __forceinline__ v8f wmma4(v2f a, v2f b, v8f c) {
    return __builtin_amdgcn_wmma_f32_16x16x4_f32(false, a, false, b, (short)0, c,
                                                 false, false);
}

// ---------------- DFT coefficient tables ----------------
// Cf[k][n] =  cos(2*pi*k*n/N)/sqrt(N)   (k >= FB -> 0)
// Sf[k][n] = -sin(2*pi*k*n/N)/sqrt(N)
// Ci[n][k] =  w_k*cos(2*pi*k*n/N)/sqrt(N),  w_0=w_{N/2}=1, else 2  (k >= FB -> 0)
// Si[n][k] = -w_k*sin(2*pi*k*n/N)/sqrt(N)
__global__ void k_tables(float* __restrict__ Cf, float* __restrict__ Sf,
                         float* __restrict__ Ci, float* __restrict__ Si) {
    int i = blockIdx.x * 256 + threadIdx.x;
    if (i >= KP * NN) return;
    {   // forward tables, [k][n] row-major (coalesced in n)
        int k = i / NN, n = i - k * NN;
        float s = 0.0f, c = 0.0f;
        if (k < FB) {
            int m = (k * n) % NN;                 // exact modular reduction
            __sincosf(TWO_PI * (float)m * (1.0f / NN), &s, &c);
        }
        Cf[i] = c * RSQN;
        Sf[i] = -s * RSQN;
    }
    {   // inverse tables, [n][k] row-major (coalesced in k)
        int n = i / KP, k = i - n * KP;
        float s = 0.0f, c = 0.0f, w = 0.0f;
        if (k < FB) {
            w = (k == 0 || k == FB - 1) ? 1.0f : 2.0f;
            int m = (k * n) % NN;
            __sincosf(TWO_PI * (float)m * (1.0f / NN), &s, &c);
        }
        Ci[i] = w * c * RSQN;
        Si[i] = -w * s * RSQN;
    }
}

// ---------------- LayerNorm: x(B,D,N) -> xn(B,N,D) ----------------
// one thread per (b,n); reads coalesced along n, transposes on write
__global__ void k_layernorm(const float* __restrict__ x,
                            const float* __restrict__ gamma,
                            const float* __restrict__ beta,
                            float* __restrict__ xn) {
    int t = blockIdx.x * 256 + threadIdx.x;      // 0 .. NB*NN-1
    int b = t / NN, n = t - b * NN;
    const float* xp = x + (size_t)b * DDIM * NN + n;
    float s = 0.0f, s2 = 0.0f;
    for (int d = 0; d < DDIM; ++d) {
        float v = xp[(size_t)d * NN];
        s += v; s2 += v * v;
    }
    float mu  = s * (1.0f / DDIM);
    float var = s2 * (1.0f / DDIM) - mu * mu;
    float r   = rsqrtf(var + 1e-5f);
    float* op = xn + (size_t)t * DDIM;
    for (int d = 0; d < DDIM; ++d) {
        float v = xp[(size_t)d * NN];
        op[d] = (v - mu) * r * gamma[d] + beta[d];
    }
}

// ---------------- context = sum_n xn / N (partial sums + atomics) ----------------
__global__ void k_ctx_reduce(const float* __restrict__ xn, float* __restrict__ ctx) {
    int b = blockIdx.y, chunk = blockIdx.x, d = threadIdx.x;
    const float* p = xn + ((size_t)b * NN + (size_t)chunk * 64) * DDIM + d;
    float s = 0.0f;
    for (int j = 0; j < 64; ++j) s += p[(size_t)j * DDIM];
    atomicAdd(&ctx[b * DDIM + d], s);
}

// ---------------- hmid = gelu(ctx/N @ W1 + b1), (32 x 256) ----------------
__global__ void k_hmid(const float* __restrict__ ctx, const float* __restrict__ W1,
                       const float* __restrict__ b1, float* __restrict__ hmid) {
    int b = blockIdx.x, j = threadIdx.x;
    const float* cb = ctx + b * DDIM;
    float acc = b1[j];
    for (int i = 0; i < DDIM; ++i)
        acc += (cb[i] * (1.0f / NN)) * W1[i * DDIM + j];
    hmid[b * DDIM + j] = gelu_exact(acc);
}

// ---------------- adapt = hmid @ W2 + b2, (32 x 18448) via WMMA ----------------
__global__ void k_adapt(const float* __restrict__ hmid, const float* __restrict__ W2,
                        const float* __restrict__ b2, float* __restrict__ adapt) {
    int n0 = blockIdx.x * 16;                    // 1153 col tiles (exact)
    int m0 = blockIdx.y * 16;                    // 2 row tiles
    int lane = threadIdx.x, half = lane >> 4, li = lane & 15;
    v8f acc = {0.f, 0.f, 0.f, 0.f, 0.f, 0.f, 0.f, 0.f};
    const float* ap = hmid + (size_t)(m0 + li) * DDIM + 2 * half;
    const float* bp = W2 + (size_t)(2 * half) * ADIM + n0 + li;
    for (int k0 = 0; k0 < DDIM; k0 += 4) {
        v2f a = *(const v2f*)ap;
        v2f bf; bf.x = bp[0]; bf.y = bp[ADIM];
        acc = wmma4(a, bf, acc);
        ap += 4; bp += (size_t)4 * ADIM;
    }
    float bias = b2[n0 + li];
#pragma unroll
    for (int r = 0; r < 8; ++r) {
        int row = m0 + r + 8 * half;
        adapt[(size_t)row * ADIM + n0 + li] = acc[r] + bias;
    }
}

// ---------------- forward DFT: Fr/Fi(bh,KP,32) = [Cf;Sf] @ xn_head ----------------
// one wave per (16 freq rows x 32 head channels) tile
__global__ void k_fwd(const float* __restrict__ xn, const float* __restrict__ Cf,
                      const float* __restrict__ Sf, float* __restrict__ Fr,
                      float* __restrict__ Fi) {
    int mtile = blockIdx.x;                      // 73 tiles over KP
    int bh = blockIdx.y;                         // 256 (b,h) pairs
    int b = bh >> 3, h = bh & 7;
    int lane = threadIdx.x, half = lane >> 4, li = lane & 15;
    int row = mtile * 16 + li;                   // < KP, tables padded w/ zeros
    const float* acp = Cf + (size_t)row * NN + 2 * half;
    const float* asp = Sf + (size_t)row * NN + 2 * half;
    const float* bp  = xn + ((size_t)b * NN + 2 * half) * DDIM + h * HD + li;
    v8f aR0 = {0,0,0,0,0,0,0,0}, aR1 = {0,0,0,0,0,0,0,0};
    v8f aI0 = {0,0,0,0,0,0,0,0}, aI1 = {0,0,0,0,0,0,0,0};
    for (int k0 = 0; k0 < NN; k0 += 4) {
        v2f ac = *(const v2f*)acp;
        v2f as = *(const v2f*)asp;
        v2f b0, b1;
        b0.x = bp[0];     b0.y = bp[DDIM];
        b1.x = bp[16];    b1.y = bp[DDIM + 16];
        aR0 = wmma4(ac, b0, aR0);
        aR1 = wmma4(ac, b1, aR1);
        aI0 = wmma4(as, b0, aI0);
        aI1 = wmma4(as, b1, aI1);
        acp += 4; asp += 4; bp += (size_t)4 * DDIM;
    }
    size_t base = ((size_t)bh * KP + mtile * 16 + 8 * half) * HD + li;
    float* fr = Fr + base;
    float* fi = Fi + base;
#pragma unroll
    for (int r = 0; r < 8; ++r) {
        fr[(size_t)r * HD]      = aR0[r];
        fr[(size_t)r * HD + 16] = aR1[r];
        fi[(size_t)r * HD]      = aI0[r];
        fi[(size_t)r * HD + 16] = aI1[r];
    }
}

// ---------------- spectral modulation + magnitude GELU (pointwise) ----------------
__global__ void k_mod(float* __restrict__ Fr, float* __restrict__ Fi,
                      const float* __restrict__ adapt,
                      const float* __restrict__ bfilt,
                      const float* __restrict__ bbias) {
    int bh = blockIdx.y;
    int k  = blockIdx.x * 8 + (threadIdx.x >> 5);
    int c  = threadIdx.x & 31;
    if (k >= FB) return;
    int b = bh >> 3, h = bh & 7;
    size_t fidx = ((size_t)bh * KP + k) * HD + c;
    float fr = Fr[fidx], fi = Fi[fidx];
    size_t aidx = (size_t)b * ADIM + (size_t)(h * FB + k) * 2;
    float ef = bfilt[h * FB + k] * (1.0f + adapt[aidx]);
    float eb = bbias[h * FB + k] + adapt[aidx + 1];
    fr = fr * ef + eb;
    fi = fi * ef;
    float mag = sqrtf(fr * fr + fi * fi);
    float g = gelu_exact(mag) / (mag + 1e-6f);
    Fr[fidx] = fr * g;
    Fi[fidx] = fi * g;
}

// ---------------- inverse DFT + residual: out(B,D,N) = x + Ci@Fr + Si@Fi ----------------
__global__ void k_inv(const float* __restrict__ Fr, const float* __restrict__ Fi,
                      const float* __restrict__ Ci, const float* __restrict__ Si,
                      const float* __restrict__ x, float* __restrict__ out) {
    int mtile = blockIdx.x;                      // 144 tiles over N
    int bh = blockIdx.y;
    int b = bh >> 3, h = bh & 7;
    int lane = threadIdx.x, half = lane >> 4, li = lane & 15;
    int row = mtile * 16 + li;                   // n index for A rows
    const float* acp = Ci + (size_t)row * KP + 2 * half;
    const float* asp = Si + (size_t)row * KP + 2 * half;
    const float* frp = Fr + ((size_t)bh * KP + 2 * half) * HD + li;
    const float* fip = Fi + ((size_t)bh * KP + 2 * half) * HD + li;
    v8f a0 = {0,0,0,0,0,0,0,0}, a1 = {0,0,0,0,0,0,0,0};
    for (int k0 = 0; k0 < KP; k0 += 4) {
        v2f ac = *(const v2f*)acp;
        v2f as = *(const v2f*)asp;
        v2f br0, br1, bi0, bi1;
        br0.x = frp[0];   br0.y = frp[HD];
        br1.x = frp[16];  br1.y = frp[HD + 16];
        bi0.x = fip[0];   bi0.y = fip[HD];
        bi1.x = fip[16];  bi1.y = fip[HD + 16];
        a0 = wmma4(ac, br0, a0);
        a1 = wmma4(ac, br1, a1);
        a0 = wmma4(as, bi0, a0);
        a1 = wmma4(as, bi1, a1);
        acp += 4; asp += 4; frp += 4 * HD; fip += 4 * HD;
    }
    int n  = mtile * 16 + 8 * half;
    int d0 = h * HD + li;
    const float* xb = x   + ((size_t)b * DDIM + d0) * NN + n;
    float*       ob = out + ((size_t)b * DDIM + d0) * NN + n;
#pragma unroll
    for (int r = 0; r < 8; ++r) {
        ob[r] = xb[r] + a0[r];
        ob[(size_t)16 * NN + r] = xb[(size_t)16 * NN + r] + a1[r];
    }
}

extern "C" void kernel_launch(void* const* d_in, const int* in_sizes, int n_in,
                              void* d_out, int out_size, void* d_ws, size_t ws_size,
                              hipStream_t stream) {
    (void)in_sizes; (void)n_in; (void)out_size; (void)ws_size;
    const float* x     = (const float*)d_in[0];
    const float* bfilt = (const float*)d_in[1];
    const float* bbias = (const float*)d_in[2];
    const float* gamma = (const float*)d_in[3];
    const float* beta  = (const float*)d_in[4];
    const float* W1    = (const float*)d_in[5];
    const float* b1    = (const float*)d_in[6];
    const float* W2    = (const float*)d_in[7];
    const float* b2    = (const float*)d_in[8];

    float* out = (float*)d_out;
    float* ws  = (float*)d_ws;
    float* Cf = ws + OFF_CF;
    float* Sf = ws + OFF_SF;
    float* Ci = ws + OFF_CI;
    float* Si = ws + OFF_SI;
    float* Fr = ws + OFF_FR;
    float* Fi = ws + OFF_FI;
    float* ctx  = ws + OFF_CTX;
    float* hmid = ws + OFF_HMID;
    float* adpt = ws + OFF_ADPT;
    float* xn = out;   // stage x_norm in d_out; dead before k_inv overwrites it

    k_tables<<<(KP * NN) / 256, 256, 0, stream>>>(Cf, Sf, Ci, Si);
    k_layernorm<<<(NB * NN) / 256, 256, 0, stream>>>(x, gamma, beta, xn);
    hipMemsetAsync(ctx, 0, (size_t)NB * DDIM * sizeof(float), stream);
    k_ctx_reduce<<<dim3(NN / 64, NB), 256, 0, stream>>>(xn, ctx);
    k_hmid<<<NB, DDIM, 0, stream>>>(ctx, W1, b1, hmid);
    k_adapt<<<dim3(ADIM / 16, NB / 16), 32, 0, stream>>>(hmid, W2, b2, adpt);
    k_fwd<<<dim3(KP / 16, NB * HEADS), 32, 0, stream>>>(xn, Cf, Sf, Fr, Fi);
    k_mod<<<dim3((FB + 7) / 8, NB * HEADS), 256, 0, stream>>>(Fr, Fi, adpt, bfilt, bbias);
    k_inv<<<dim3(NN / 16, NB * HEADS), 32, 0, stream>>>(Fr, Fi, Ci, Si, x, out);
}